// PETValidatorContext_13211319402654
// MI455X (gfx1250) — compile-verified
//
#include <hip/hip_runtime.h>

// ---------------------------------------------------------------------------
// PET GIN network for MI455X (gfx1250), wave32.
//   conv3d (VALU, weights in LDS) -> 4x [ z = h + scatter_add(h[src]->dst) ;
//   h = relu(z@W1+b1)@W2+b2 (fused dual-GEMM via V_WMMA_F32_16X16X4_F32) ]
// ---------------------------------------------------------------------------

typedef float v2f __attribute__((ext_vector_type(2)));
typedef float v8f __attribute__((ext_vector_type(8)));

#define N_NODES 32768
#define N_EDGES 524288

// ---------------------------------------------------------------------------
// Conv3d: X [3,34,34,34] * W [64,3,3,3,3] + B -> H node-major [32768, 64]
// ---------------------------------------------------------------------------
__global__ __launch_bounds__(256) void conv3d_embed(
    const float* __restrict__ X, const float* __restrict__ W,
    const float* __restrict__ B, float* __restrict__ H)
{
    __shared__ float wl[64 * 81];   // 20.7 KB
    for (int j = threadIdx.x; j < 64 * 81; j += 256) wl[j] = W[j];
    __syncthreads();

    const int gid  = blockIdx.x * 256 + threadIdx.x;   // exact cover 32768*64
    const int c    = gid & 63;
    const int node = gid >> 6;
    const int w = node & 31, h = (node >> 5) & 31, d = node >> 10;

    float sum = B[c];
    const float* wp = wl + c * 81;
#pragma unroll
    for (int i = 0; i < 3; ++i)
#pragma unroll
        for (int kd = 0; kd < 3; ++kd)
#pragma unroll
            for (int kh = 0; kh < 3; ++kh)
#pragma unroll
                for (int kw = 0; kw < 3; ++kw)
                    sum = fmaf(X[(((size_t)i * 34 + d + kd) * 34 + h + kh) * 34 + (w + kw)],
                               wp[((i * 3 + kd) * 3 + kh) * 3 + kw], sum);
    H[(size_t)node * 64 + c] = sum;
}

// ---------------------------------------------------------------------------
// z = h  (float4 copy); scatter-add then accumulates agg on top -> z = h + agg
// ---------------------------------------------------------------------------
__global__ void copy_f4(const float* __restrict__ in, float* __restrict__ out, int n4)
{
    int i = blockIdx.x * blockDim.x + threadIdx.x;
    if (i < n4) ((float4*)out)[i] = ((const float4*)in)[i];
}

// ---------------------------------------------------------------------------
// z[dst] += h[src]; DIM/4 threads per edge, float4 gather + 4 f32 atomics
// ---------------------------------------------------------------------------
template<int DIM>
__global__ void scatter_add(const float* __restrict__ H, const int* __restrict__ src,
                            const int* __restrict__ dst, float* __restrict__ Z, int nEdges)
{
    constexpr int TPE = DIM / 4;
    long long tid = (long long)blockIdx.x * blockDim.x + threadIdx.x;
    int e = (int)(tid / TPE);
    if (e >= nEdges) return;
    int c = (int)(tid % TPE) * 4;

    int s = src[e], d = dst[e];
    const float4 v = *(const float4*)(H + (size_t)s * DIM + c);
    float* zp = Z + (size_t)d * DIM + c;
    atomicAdd(zp + 0, v.x);
    atomicAdd(zp + 1, v.y);
    atomicAdd(zp + 2, v.z);
    atomicAdd(zp + 3, v.w);
}

// ---------------------------------------------------------------------------
// Fused GIN MLP:  Out = [relu] ( relu(Z @ W1 + b1) @ W2 + b2 )
//   Z: [N, K1], W1: [K1, M1], W2: [M1, M2]  (row-major, f32)
//   128 threads = 4 waves; wave owns 16 rows; block covers 64 rows.
//   Both GEMMs via V_WMMA_F32_16X16X4_F32; intermediate T in wave-private LDS.
//   No divergence: EXEC is all-1s for every WMMA (ISA requirement).
// ---------------------------------------------------------------------------
template<int K1, int M1, int M2, bool FINAL_RELU>
__global__ __launch_bounds__(128) void gin_mlp(
    const float* __restrict__ Z,  const float* __restrict__ W1,
    const float* __restrict__ B1, const float* __restrict__ W2,
    const float* __restrict__ B2, float* __restrict__ Out)
{
    constexpr int TS = M1 + 4;                  // LDS row stride (bank-conflict free)
    __shared__ float Tlds[4][16 * TS];          // <= 33.8 KB

    const int wave  = threadIdx.x >> 5;
    const int lane  = threadIdx.x & 31;
    const int lhalf = lane >> 4;                // 0: K/M low half, 1: high half
    const int l16   = lane & 15;
    const int row0  = blockIdx.x * 64 + wave * 16;

    // ---------------- Stage 1: T = relu(Z @ W1 + b1) ----------------
    v8f acc1[M1 / 16] = {};
    const float* zrow = Z + (size_t)(row0 + l16) * K1;
#pragma unroll
    for (int k = 0; k < K1; k += 4) {
        // A frag: (row = row0+l16, k + 2*lhalf .. +1)   [ISA 16x4 f32 layout]
        v2f a = *(const v2f*)(zrow + k + 2 * lhalf);
#pragma unroll
        for (int ct = 0; ct < M1 / 16; ++ct) {
            const int col = ct * 16 + l16;
            v2f b;
            b.x = W1[(size_t)(k + 2 * lhalf) * M1 + col];
            b.y = W1[(size_t)(k + 1 + 2 * lhalf) * M1 + col];
            acc1[ct] = __builtin_amdgcn_wmma_f32_16x16x4_f32(
                false, a, false, b, (short)0, acc1[ct], false, false);
        }
    }
    float* T = &Tlds[wave][0];
#pragma unroll
    for (int ct = 0; ct < M1 / 16; ++ct) {
        const int col = ct * 16 + l16;
        const float bv = B1[col];
#pragma unroll
        for (int r = 0; r < 8; ++r) {
            float v = acc1[ct][r] + bv;
            v = v > 0.f ? v : 0.f;                       // relu
            T[(r + 8 * lhalf) * TS + col] = v;           // C-layout -> row-major
        }
    }
    __syncthreads();   // LDS visibility (wave-private data, but cheap & safe)

    // ---------------- Stage 2: Out = T @ W2 + b2 ----------------
    v8f acc2[M2 / 16] = {};
#pragma unroll
    for (int k = 0; k < M1; k += 4) {
        v2f a = *(const v2f*)(T + l16 * TS + k + 2 * lhalf);
#pragma unroll
        for (int ct = 0; ct < M2 / 16; ++ct) {
            const int col = ct * 16 + l16;
            v2f b;
            b.x = W2[(size_t)(k + 2 * lhalf) * M2 + col];
            b.y = W2[(size_t)(k + 1 + 2 * lhalf) * M2 + col];
            acc2[ct] = __builtin_amdgcn_wmma_f32_16x16x4_f32(
                false, a, false, b, (short)0, acc2[ct], false, false);
        }
    }
#pragma unroll
    for (int ct = 0; ct < M2 / 16; ++ct) {
        const int col = ct * 16 + l16;
        const float bv = B2[col];
#pragma unroll
        for (int r = 0; r < 8; ++r) {
            float v = acc2[ct][r] + bv;
            if (FINAL_RELU) v = v > 0.f ? v : 0.f;
            Out[(size_t)(row0 + r + 8 * lhalf) * M2 + col] = v;
        }
    }
}

// ---------------------------------------------------------------------------
extern "C" void kernel_launch(void* const* d_in, const int* in_sizes, int n_in,
                              void* d_out, int out_size, void* d_ws, size_t ws_size,
                              hipStream_t stream)
{
    const float* pet    = (const float*)d_in[0];
    const int*   idx    = (const int*)d_in[1];        // [2, N_EDGES]
    /* d_in[2] = pet_scan_attn: dropped by PyG GIN (edge_weight unused)     */
    const float* conv_w = (const float*)d_in[3];
    const float* conv_b = (const float*)d_in[4];
    // mlp_params flattened: layer L tensors at d_in[5 + 4*L + {0..3}] = w1,b1,w2,b2
    const float* W1[4]; const float* B1[4]; const float* W2[4]; const float* B2[4];
    for (int L = 0; L < 4; ++L) {
        W1[L] = (const float*)d_in[5 + 4 * L + 0];
        B1[L] = (const float*)d_in[5 + 4 * L + 1];
        W2[L] = (const float*)d_in[5 + 4 * L + 2];
        B2[L] = (const float*)d_in[5 + 4 * L + 3];
    }
    const int* src = idx;
    const int* dst = idx + N_EDGES;

    float* hbuf = (float*)d_ws;                 // 32768*128 floats = 16 MB
    float* zbuf = hbuf + (size_t)N_NODES * 128; // 16 MB
    float* out  = (float*)d_out;                // 32768*64 floats

    // conv embedding -> hbuf (dim 64)
    conv3d_embed<<<(N_NODES * 64) / 256, 256, 0, stream>>>(pet, conv_w, conv_b, hbuf);

    // ---- layer 1: dim 64 -> 128 ----
    copy_f4<<<(N_NODES * 64 / 4 + 255) / 256, 256, 0, stream>>>(hbuf, zbuf, N_NODES * 64 / 4);
    scatter_add<64><<<(N_EDGES * 16 + 255) / 256, 256, 0, stream>>>(hbuf, src, dst, zbuf, N_EDGES);
    gin_mlp<64, 128, 128, true><<<N_NODES / 64, 128, 0, stream>>>(
        zbuf, W1[0], B1[0], W2[0], B2[0], hbuf);

    // ---- layers 2,3: dim 128 -> 128 ----
    for (int L = 1; L <= 2; ++L) {
        copy_f4<<<(N_NODES * 128 / 4 + 255) / 256, 256, 0, stream>>>(hbuf, zbuf, N_NODES * 128 / 4);
        scatter_add<128><<<(N_EDGES * 32 + 255) / 256, 256, 0, stream>>>(hbuf, src, dst, zbuf, N_EDGES);
        gin_mlp<128, 128, 128, true><<<N_NODES / 64, 128, 0, stream>>>(
            zbuf, W1[L], B1[L], W2[L], B2[L], hbuf);
    }

    // ---- layer 4: dim 128 -> 64, no final relu, straight to d_out ----
    copy_f4<<<(N_NODES * 128 / 4 + 255) / 256, 256, 0, stream>>>(hbuf, zbuf, N_NODES * 128 / 4);
    scatter_add<128><<<(N_EDGES * 32 + 255) / 256, 256, 0, stream>>>(hbuf, src, dst, zbuf, N_EDGES);
    gin_mlp<128, 64, 64, false><<<N_NODES / 64, 128, 0, stream>>>(
        zbuf, W1[3], B1[3], W2[3], B2[3], out);

    (void)in_sizes; (void)n_in; (void)out_size; (void)ws_size;
}